// ProtoNet_47158740910868
// MI455X (gfx1250) — compile-verified
//
#include <hip/hip_runtime.h>
#include <cmath>

// ---------------- problem constants ----------------
constexpr int   WAY = 5, SHOT = 5, RERANK = 10;
constexpr int   DIM = 1024, NQ = 8192;
constexpr float SCALE = 32.0f;          // sqrt(DIM)
constexpr float MIN_NORM = 1e-5f;
constexpr int   NPAD = 16;              // protos padded to 16 columns for WMMA

// ---------------- workspace layout (float offsets) ----------------
constexpr int OFF_COLSUM = 0;           // 1024   : column sums of data_query
constexpr int OFF_ALLD   = 1024;        // 1024   : all_data mean vector
constexpr int OFF_X1     = 2048;        // 5*1024 : x1
constexpr int OFF_H1     = 7168;        // 5*128
constexpr int OFF_H2     = 7808;        // 5*64
constexpr int OFF_C      = 8128;        // c[5]
constexpr int OFF_P2     = 8144;        // ||proto_i||^2
constexpr int OFF_TP2    = 8160;        // ||test_proto_i||^2
constexpr int OFF_SW     = 8176;        // sw[5][5]
constexpr int OFF_QN     = 8704;        // 8192   : ||q||^2 per query row
constexpr int OFF_DIS    = 16896;       // 5*8192 : dis_mat
constexpr int OFF_PT     = 57856;       // [DIM][16] protos (transposed, zero padded)
constexpr int OFF_TT     = 74240;       // [DIM][16] test protos
constexpr int WS_FLOATS  = 90624;

typedef __attribute__((ext_vector_type(2))) float v2f;
typedef __attribute__((ext_vector_type(8))) float v8f;

// ---------------- scalar helpers ----------------
// e2p(v,c) = g * v  with g depending only on ||v||^2 (expmap0 + project)
__device__ __forceinline__ float e2p_scale(float n2, float c) {
  float sc  = sqrtf(c);
  float n   = sqrtf(fmaxf(n2, 0.f));
  float ncl = fmaxf(n, MIN_NORM);
  float t   = tanhf(sc * ncl);
  float g   = t / (sc * ncl);
  float nh  = fmaxf(g * n, MIN_NORM);       // norm of mapped vector (clipped)
  float mx  = 0.999f / sc;                  // (1 - 0.001)/sqrt(c)
  if (nh > mx) g *= mx / nh;
  return g;
}

// dist of mobius_add(-x, y): needs only x2=||x||^2, y2=||y||^2, xy=(-x).y
__device__ __forceinline__ float hyp_dist(float x2, float y2, float xy, float c) {
  float A    = 1.f + 2.f * c * xy + c * y2;
  float B    = 1.f - c * x2;
  float num2 = A * A * x2 + 2.f * A * B * xy + B * B * y2;
  float den  = fmaxf(1.f + 2.f * c * xy + c * c * x2 * y2, MIN_NORM);
  float sc   = sqrtf(c);
  float z    = sc * sqrtf(fmaxf(num2, 0.f)) / den;
  z = fminf(fmaxf(z, -1.f + 1e-7f), 1.f - 1e-7f);
  float at = 0.5f * (log1pf(z) - log1pf(-z));
  return 2.f / sc * at;
}

__device__ __forceinline__ float sigmoidf(float x) { return 1.f / (1.f + expf(-x)); }

__device__ __forceinline__ void softmax_inplace(float* x, int n) {
  float m = x[0];
  for (int i = 1; i < n; ++i) m = fmaxf(m, x[i]);
  float s = 0.f;
  for (int i = 0; i < n; ++i) { x[i] = expf(x[i] - m); s += x[i]; }
  float inv = 1.f / s;
  for (int i = 0; i < n; ++i) x[i] *= inv;
}

// poincare_mean of rows r = e[r]*raw_r  ->  out = sum coef[r]*raw_r,
// out_n2 = ||out||^2.  G = n x n Gram of RAW rows.
__device__ void poincare_coef(const float* G, int n, const float* e, float c,
                              float* coef, float* out_n2) {
  float lam[16], al[16];
  float lsum = 0.f;
  for (int r = 0; r < n; ++r) {
    float grr = G[r * n + r];
    float x2  = e[r] * e[r] * grr;
    float a   = 2.f * e[r] / (1.f + c * x2);            // xk = a*raw
    float xk2 = a * a * grr;
    float l   = 1.f / sqrtf(fmaxf(1.f - c * xk2, 1e-7f));
    lam[r] = l; al[r] = a; lsum += l;
  }
  float beta[16];
  for (int r = 0; r < n; ++r) beta[r] = lam[r] * al[r] / lsum;
  float mk2 = 0.f;
  for (int r = 0; r < n; ++r)
    for (int s = 0; s < n; ++s) mk2 += beta[r] * beta[s] * G[r * n + s];
  float gam = 1.f / (1.f + sqrtf(fmaxf(1.f - c * mk2, 1e-7f)));
  for (int r = 0; r < n; ++r) coef[r] = gam * beta[r];
  *out_n2 = gam * gam * mk2;
}

// ---------------- kernel: zero pad region ----------------
__global__ void k_zero(float* __restrict__ p, int n) {
  int i = blockIdx.x * blockDim.x + threadIdx.x;
  if (i < n) p[i] = 0.f;
}

// ---------------- kernel: per-row ||q||^2 (one wave32 per row) ----------------
__global__ __launch_bounds__(256) void k_rownorm(const float* __restrict__ Q,
                                                 float* __restrict__ qn2) {
  const int wave = threadIdx.x >> 5, lane = threadIdx.x & 31;
  const int row = blockIdx.x * 8 + wave;
  const float* r = Q + (size_t)row * DIM;
  float acc = 0.f;
  for (int k = lane; k < DIM; k += 32) { float v = r[k]; acc += v * v; }
  for (int off = 16; off > 0; off >>= 1) acc += __shfl_xor(acc, off, 32);
  if (lane == 0) qn2[row] = acc;
}

// ---------------- kernel: column sums of data_query ----------------
__global__ __launch_bounds__(256) void k_colsum(const float* __restrict__ Q,
                                                float* __restrict__ colsum) {
  const int col = blockIdx.x * 256 + threadIdx.x;
  float acc = 0.f;
  for (int r = 0; r < NQ; ++r) acc += Q[(size_t)r * DIM + col];
  colsum[col] = acc;
}

// ---------------- kernel: all_data, c[5], per-way protos (one block) ----------------
__global__ __launch_bounds__(256) void k_setup(
    const float* __restrict__ shot,
    const float* __restrict__ Wc1a, const float* __restrict__ Wc1b, const float* __restrict__ bc1,
    const float* __restrict__ Wc2,  const float* __restrict__ bc2,
    const float* __restrict__ Wc3,  const float* __restrict__ bc3,
    const float* __restrict__ Ws1,  const float* __restrict__ bs1,
    const float* __restrict__ Ws2,  const float* __restrict__ bs2,
    float* __restrict__ ws) {
  const int tid = threadIdx.x;
  float* colsum = ws + OFF_COLSUM;
  float* alld   = ws + OFF_ALLD;
  float* x1     = ws + OFF_X1;
  float* h1     = ws + OFF_H1;
  float* h2     = ws + OFF_H2;
  float* cvec   = ws + OFF_C;
  float* p2     = ws + OFF_P2;
  float* swp    = ws + OFF_SW;
  float* Pt     = ws + OFF_PT;

  __shared__ float G[25];
  __shared__ float coef5[5];

  // all_data = mean over all 25+8192 rows
  for (int cc = tid; cc < DIM; cc += 256) {
    float s = colsum[cc];
    for (int r = 0; r < SHOT * WAY; ++r) s += shot[r * DIM + cc];
    alld[cc] = s / (float)(SHOT * WAY + NQ);
  }
  __threadfence_block(); __syncthreads();

  // x1 = (all_data*(WAY+NQ) - mean_proto*SHOT) / (WAY+NQ-SHOT)
  for (int idx = tid; idx < WAY * DIM; idx += 256) {
    int w = idx / DIM, cc = idx % DIM;
    float mp = 0.f;
    for (int s = 0; s < SHOT; ++s) mp += shot[(s * WAY + w) * DIM + cc];
    mp *= (1.f / SHOT);
    x1[idx] = (alld[cc] * (float)(WAY + NQ) - mp * (float)SHOT) / (float)(WAY + NQ - SHOT);
  }
  __threadfence_block(); __syncthreads();

  // h1 = relu(x1@Wc1a + all_data@Wc1b + bc1)   (5x128)
  for (int idx = tid; idx < WAY * 128; idx += 256) {
    int w = idx / 128, j = idx % 128;
    float acc = bc1[j];
    for (int k = 0; k < DIM; ++k)
      acc += x1[w * DIM + k] * Wc1a[k * 128 + j] + alld[k] * Wc1b[k * 128 + j];
    h1[idx] = fmaxf(acc, 0.f);
  }
  __threadfence_block(); __syncthreads();

  // h2 = relu(h1@Wc2 + bc2)   (5x64)
  for (int idx = tid; idx < WAY * 64; idx += 256) {
    int w = idx / 64, j = idx % 64;
    float acc = bc2[j];
    for (int k = 0; k < 128; ++k) acc += h1[w * 128 + k] * Wc2[k * 64 + j];
    h2[idx] = fmaxf(acc, 0.f);
  }
  __threadfence_block(); __syncthreads();

  // c = sigmoid(h2@Wc3 + bc3) * L / DIVIDE
  if (tid < WAY) {
    float acc = bc3[0];
    for (int k = 0; k < 64; ++k) acc += h2[tid * 64 + k] * Wc3[k];
    cvec[tid] = sigmoidf(acc) * 0.1f;     // L=1, DIVIDE=10
  }
  __threadfence_block(); __syncthreads();

  // per-way protos.  All hyperbolic ops collapse to scalar coefficients on
  // the 5 raw shot rows given the 5x5 Gram matrix.
  for (int i = 0; i < WAY; ++i) {
    if (tid < 15) {  // upper-triangular Gram of raw rows
      int a = 0, b = 0, cnt = tid;
      for (a = 0; a < SHOT; ++a) { int rl = SHOT - a; if (cnt < rl) { b = a + cnt; break; } cnt -= rl; }
      const float* ra = shot + (size_t)(a * WAY + i) * DIM;
      const float* rb = shot + (size_t)(b * WAY + i) * DIM;
      float d = 0.f;
      for (int k = 0; k < DIM; ++k) d += ra[k] * rb[k];
      G[a * 5 + b] = d; G[b * 5 + a] = d;
    }
    __syncthreads();
    if (tid == 0) {
      float ci = cvec[i];
      float g[5];
      for (int s = 0; s < SHOT; ++s) g[s] = e2p_scale(G[s * 5 + s], ci);
      float sdm[25];
      for (int a = 0; a < SHOT; ++a)
        for (int b = 0; b < SHOT; ++b) {
          float x2 = g[a] * g[a] * G[a * 5 + a];
          float y2 = g[b] * g[b] * G[b * 5 + b];
          float xy = -g[a] * g[b] * G[a * 5 + b];
          sdm[a * 5 + b] = -hyp_dist(x2, y2, xy, ci) / SCALE;   // pre-negated for softmax
        }
      softmax_inplace(sdm, 25);
      float hh[SHOT];
      for (int j = 0; j < SHOT; ++j) {
        float a = bs1[j];
        for (int p = 0; p < 25; ++p) a += sdm[p] * Ws1[p * SHOT + j];
        hh[j] = fmaxf(a, 0.f);
      }
      float oo[SHOT];
      for (int k = 0; k < SHOT; ++k) {
        float a = bs2[k];
        for (int j = 0; j < SHOT; ++j) a += hh[j] * Ws2[j * SHOT + k];
        oo[k] = a;
      }
      softmax_inplace(oo, SHOT);                 // sw
      float e[5];
      for (int s = 0; s < SHOT; ++s) {
        swp[i * SHOT + s] = oo[s];
        float w = (float)SHOT * oo[s];           // sw_i = sw*SHOT
        e[s] = w * e2p_scale(w * w * G[s * 5 + s], ci);
      }
      float p2i;
      poincare_coef(G, SHOT, e, ci, coef5, &p2i);
      p2[i] = p2i;
    }
    __threadfence_block(); __syncthreads();
    for (int cc = tid; cc < DIM; cc += 256) {
      float s = 0.f;
      for (int st = 0; st < SHOT; ++st) s += coef5[st] * shot[(st * WAY + i) * DIM + cc];
      Pt[cc * NPAD + i] = s;
    }
    __syncthreads();
  }
}

// ---------------- kernel: Q @ P via V_WMMA_F32_16X16X4_F32, fused distance ----------------
// Each wave owns a 16-row x 16-col tile (5 cols valid).  A tiles staged per
// wave in LDS (coalesced), B chunks staged block-wide.  mode 0 -> dis_mat,
// mode 1 -> final output (-dist).
__global__ __launch_bounds__(256) void k_gemm_dist(
    const float* __restrict__ Q, const float* __restrict__ Pm,
    const float* __restrict__ qn2, const float* __restrict__ cv,
    const float* __restrict__ x2v, float* __restrict__ disOut,
    float* __restrict__ finalOut, int mode) {
  constexpr int KCH = 64;
  __shared__ float sB[KCH * NPAD];        // 4 KB
  __shared__ float sA[8][16 * KCH];       // 32 KB (4 KB per wave)
  const int tid  = threadIdx.x;
  const int wave = tid >> 5, lane = tid & 31;
  const int row0 = (blockIdx.x * 8 + wave) * 16;
  const int ln15 = lane & 15;             // M for A, N for B/D
  const int koff = (lane >> 4) << 1;      // K offset per ISA A/B layout
  float* saw = sA[wave];
  const float* qbase = Q + (size_t)row0 * DIM;
  v8f acc = {};

  for (int k0 = 0; k0 < DIM; k0 += KCH) {
    __syncthreads();
    for (int t = tid; t < KCH * NPAD; t += 256) sB[t] = Pm[k0 * NPAD + t];
    #pragma unroll 4
    for (int r = 0; r < 16; ++r) {
      saw[r * KCH + lane]      = qbase[(size_t)r * DIM + k0 + lane];
      saw[r * KCH + lane + 32] = qbase[(size_t)r * DIM + k0 + lane + 32];
    }
    if (k0 + KCH < DIM)  // pull next chunk toward L2 (global_prefetch_b8)
      __builtin_prefetch(qbase + (size_t)ln15 * DIM + k0 + KCH, 0, 1);
    __syncthreads();
    #pragma unroll
    for (int kk = 0; kk < KCH; kk += 4) {
      v2f a, b;
      a.x = saw[ln15 * KCH + kk + koff];
      a.y = saw[ln15 * KCH + kk + koff + 1];
      b.x = sB[(kk + koff) * NPAD + ln15];
      b.y = sB[(kk + koff + 1) * NPAD + ln15];
      acc = __builtin_amdgcn_wmma_f32_16x16x4_f32(false, a, false, b,
                                                  (short)0, acc, false, false);
    }
  }

  // epilogue: D tile -> hyperbolic distances
  if (ln15 < WAY) {
    const int nn = ln15;
    const float cc = cv[nn];
    const float x2 = x2v[nn];
    const int mbase = (lane >> 4) * 8;
    #pragma unroll
    for (int v = 0; v < 8; ++v) {
      const int q = row0 + mbase + v;
      const float n2 = qn2[q];
      const float g  = e2p_scale(n2, cc);        // e2p(query) scalar
      const float y2 = g * g * n2;
      const float xy = -g * acc[v];              // (-proto).qh
      const float d  = hyp_dist(x2, y2, xy, cc);
      if (mode == 0) disOut[nn * NQ + q] = d;
      else           finalOut[(size_t)q * WAY + nn] = -d;   // TEMP = 1
    }
  }
}

// ---------------- kernel: top-10 + rerank MLP + test protos (1 block / way) ----------------
__global__ __launch_bounds__(256) void k_rerank(
    const float* __restrict__ shot, const float* __restrict__ query,
    const float* __restrict__ Wr1, const float* __restrict__ br1,
    const float* __restrict__ Wr2, const float* __restrict__ br2,
    float* __restrict__ ws) {
  const int i = blockIdx.x;
  const int tid = threadIdx.x;
  float* dis = ws + OFF_DIS;
  float* Tt  = ws + OFF_TT;
  const float* cvec = ws + OFF_C;

  __shared__ float redv[256];
  __shared__ int   redi[256];
  __shared__ float topv[RERANK];
  __shared__ int   topi[RERANK];
  __shared__ float G[15 * 15];
  __shared__ float coef[15];
  __shared__ float rs[WAY];

  // --- top-10 by iterative argmin (yields ascending order, stable ties) ---
  for (int r = 0; r < RERANK; ++r) {
    float bv = 0.f; int bi = -1;
    for (int j = tid; j < NQ; j += 256) {
      bool used = false;
      for (int t = 0; t < r; ++t) if (topi[t] == j) used = true;
      if (used) continue;
      float v = dis[i * NQ + j];
      if (bi < 0 || v < bv) { bv = v; bi = j; }
    }
    redv[tid] = bv; redi[tid] = bi;
    __syncthreads();
    for (int s = 128; s > 0; s >>= 1) {
      if (tid < s) {
        float v2 = redv[tid + s]; int i2 = redi[tid + s];
        float v1 = redv[tid];     int i1 = redi[tid];
        if (i2 >= 0 && (i1 < 0 || v2 < v1 || (v2 == v1 && i2 < i1))) {
          redv[tid] = v2; redi[tid] = i2;
        }
      }
      __syncthreads();
    }
    if (tid == 0) { topv[r] = redv[0]; topi[r] = redi[0]; }
    __syncthreads();
  }

  // --- row sums of dis_mat rows j<=i (rows j>i are zero in the reference) ---
  for (int j = 0; j <= i; ++j) {
    float p = 0.f;
    for (int t = tid; t < NQ; t += 256) p += dis[j * NQ + t];
    redv[tid] = p; __syncthreads();
    for (int s = 128; s > 0; s >>= 1) { if (tid < s) redv[tid] += redv[tid + s]; __syncthreads(); }
    if (tid == 0) rs[j] = redv[0];
    __syncthreads();
  }

  // --- 15x15 Gram of raw rows: 10 top queries + 5 shots of this way ---
  if (tid < 120) {
    int a = 0, b = 0, cnt = tid;
    for (a = 0; a < 15; ++a) { int rl = 15 - a; if (cnt < rl) { b = a + cnt; break; } cnt -= rl; }
    const float* ra = (a < RERANK) ? query + (size_t)topi[a] * DIM
                                   : shot + (size_t)((a - RERANK) * WAY + i) * DIM;
    const float* rb = (b < RERANK) ? query + (size_t)topi[b] * DIM
                                   : shot + (size_t)((b - RERANK) * WAY + i) * DIM;
    float d = 0.f;
    for (int k = 0; k < DIM; ++k) d += ra[k] * rb[k];
    G[a * 15 + b] = d; G[b * 15 + a] = d;
  }
  __syncthreads();

  if (tid == 0) {
    const float ci = cvec[i];
    float topsum = 0.f;
    for (int k = 0; k < RERANK; ++k) topsum += topv[k];
    float o_i_d = (rs[i] - topsum) / (float)(NQ - RERANK);
    float nod[RERANK];
    for (int k = 0; k < RERANK; ++k) {
      float s = 0.f;
      for (int j = 0; j < i; ++j) s += dis[j * NQ + topi[k]];
      nod[k] = s / (float)(WAY - 1);
    }
    float o_o_d = 0.f;
    for (int j = 0; j < i; ++j) {
      float ts = 0.f;
      for (int k = 0; k < RERANK; ++k) ts += dis[j * NQ + topi[k]];
      o_o_d += rs[j] - ts;
    }
    o_o_d /= (float)((WAY - 1) * (NQ - RERANK));

    float rr[2 * RERANK + 2];
    for (int k = 0; k < RERANK; ++k) rr[k] = -topv[k] / SCALE;
    softmax_inplace(rr, RERANK);
    for (int k = 0; k < RERANK; ++k) rr[RERANK + k] = -nod[k] / SCALE;
    softmax_inplace(rr + RERANK, RERANK);
    rr[2 * RERANK]     = o_i_d;
    rr[2 * RERANK + 1] = o_o_d;

    float hh[RERANK];
    for (int j = 0; j < RERANK; ++j) {
      float a = br1[j];
      for (int p = 0; p < 2 * RERANK + 2; ++p) a += rr[p] * Wr1[p * RERANK + j];
      hh[j] = fmaxf(a, 0.f);
    }
    float ro[RERANK + 1];
    for (int k = 0; k < RERANK + 1; ++k) {
      float a = br2[k];
      for (int j = 0; j < RERANK; ++j) a += hh[j] * Wr2[j * (RERANK + 1) + k];
      ro[k] = a;
    }
    float gate = sigmoidf(ro[RERANK]);
    softmax_inplace(ro, RERANK);                    // i_weight
    float e[15];
    for (int k = 0; k < RERANK; ++k) {
      float w = ro[k] * gate * (float)(RERANK + SHOT);
      e[k] = w * e2p_scale(w * w * G[k * 15 + k], ci);
    }
    for (int s = 0; s < SHOT; ++s) {
      float w = ws[OFF_SW + i * SHOT + s] * (1.f - gate) * (float)(SHOT + RERANK);
      int r = RERANK + s;
      e[r] = w * e2p_scale(w * w * G[r * 15 + r], ci);
    }
    float tp2i;
    poincare_coef(G, 15, e, ci, coef, &tp2i);
    ws[OFF_TP2 + i] = tp2i;
  }
  __threadfence_block(); __syncthreads();

  for (int cc = tid; cc < DIM; cc += 256) {
    float s = 0.f;
    for (int r = 0; r < RERANK; ++r) s += coef[r] * query[(size_t)topi[r] * DIM + cc];
    for (int st = 0; st < SHOT; ++st) s += coef[RERANK + st] * shot[(st * WAY + i) * DIM + cc];
    Tt[cc * NPAD + i] = s;
  }
}

// ---------------- host launcher ----------------
extern "C" void kernel_launch(void* const* d_in, const int* in_sizes, int n_in,
                              void* d_out, int out_size, void* d_ws, size_t ws_size,
                              hipStream_t stream) {
  const float* shot  = (const float*)d_in[0];
  const float* query = (const float*)d_in[1];
  const float* Wc1a  = (const float*)d_in[2];
  const float* Wc1b  = (const float*)d_in[3];
  const float* bc1   = (const float*)d_in[4];
  const float* Wc2   = (const float*)d_in[5];
  const float* bc2   = (const float*)d_in[6];
  const float* Wc3   = (const float*)d_in[7];
  const float* bc3   = (const float*)d_in[8];
  const float* Wr1   = (const float*)d_in[9];
  const float* br1   = (const float*)d_in[10];
  const float* Wr2   = (const float*)d_in[11];
  const float* br2   = (const float*)d_in[12];
  const float* Ws1   = (const float*)d_in[13];
  const float* bs1   = (const float*)d_in[14];
  const float* Ws2   = (const float*)d_in[15];
  const float* bs2   = (const float*)d_in[16];
  float* ws  = (float*)d_ws;
  float* out = (float*)d_out;

  // zero the padded proto matrices (cols 5..15 must stay 0 every call)
  k_zero<<<(2 * DIM * NPAD + 255) / 256, 256, 0, stream>>>(ws + OFF_PT, 2 * DIM * NPAD);
  // pass 1 over Q: row norms (wave per row) and column sums (coalesced)
  k_rownorm<<<NQ / 8, 256, 0, stream>>>(query, ws + OFF_QN);
  k_colsum<<<DIM / 256, 256, 0, stream>>>(query, ws + OFF_COLSUM);
  // tiny scalar stage: all_data, c[5], per-way protos (Gram algebra)
  k_setup<<<1, 256, 0, stream>>>(shot, Wc1a, Wc1b, bc1, Wc2, bc2, Wc3, bc3,
                                 Ws1, bs1, Ws2, bs2, ws);
  // WMMA GEMM #1 + fused hyperbolic distance -> dis_mat
  k_gemm_dist<<<NQ / 128, 256, 0, stream>>>(query, ws + OFF_PT, ws + OFF_QN,
                                            ws + OFF_C, ws + OFF_P2,
                                            ws + OFF_DIS, nullptr, 0);
  // per-way top-10 + rerank MLP + test protos
  k_rerank<<<WAY, 256, 0, stream>>>(shot, query, Wr1, br1, Wr2, br2, ws);
  // WMMA GEMM #2 + fused distance -> final output (-new_dis.T)
  k_gemm_dist<<<NQ / 128, 256, 0, stream>>>(query, ws + OFF_TT, ws + OFF_QN,
                                            ws + OFF_C, ws + OFF_TP2,
                                            nullptr, out, 1);
  (void)in_sizes; (void)n_in; (void)out_size; (void)ws_size;
}